// RugE_Loss_17652315586942
// MI455X (gfx1250) — compile-verified
//
#include <hip/hip_runtime.h>
#include <hip/hip_bf16.h>

typedef __attribute__((ext_vector_type(2))) float v2f;
typedef __attribute__((ext_vector_type(8))) float v8f;

#define POS_LEN    32768
#define NEG_TOTAL  (32768LL * 64LL)
#define N_RULES    262144
#define N_UNLABEL  131072
#define EMB_DIM    256
#define ENT_FLOATS (100000LL * 256LL)
#define REL_FLOATS (500LL * 256LL)
#define WDECAY     1e-5f

__device__ __forceinline__ float softplusf(float x) {
    // numerically stable softplus: max(x,0) + log1p(exp(-|x|))
    return fmaxf(x, 0.0f) + log1pf(__expf(-fabsf(x)));
}
__device__ __forceinline__ float sigmoidf(float x) {
    return 1.0f / (1.0f + __expf(-x));
}

__device__ __forceinline__ float block_reduce_sum(float v) {
    __shared__ float smem[32];
    int lane = threadIdx.x & 31;
    int wid  = threadIdx.x >> 5;
#pragma unroll
    for (int off = 16; off > 0; off >>= 1) v += __shfl_down(v, off, 32);
    if (lane == 0) smem[wid] = v;
    __syncthreads();
    int nw = blockDim.x >> 5;
    v = ((int)threadIdx.x < nw) ? smem[threadIdx.x] : 0.0f;
    if (wid == 0) {
#pragma unroll
        for (int off = 16; off > 0; off >>= 1) v += __shfl_down(v, off, 32);
    }
    return v;
}

// ---------------------------------------------------------------------------
// Kernel 0: zero accumulators + pi_grad scratch
// ---------------------------------------------------------------------------
__global__ void zero_ws_kernel(float* acc, float* pi_grad, int n) {
    int i = blockIdx.x * blockDim.x + threadIdx.x;
    if (i < 8) acc[i] = 0.0f;
    for (int k = i; k < n; k += gridDim.x * blockDim.x) pi_grad[k] = 0.0f;
}

// ---------------------------------------------------------------------------
// Kernel 1: positive_loss + negative_loss  (fused streaming reduction)
// ---------------------------------------------------------------------------
__global__ void pos_neg_kernel(const float* __restrict__ pos,
                               const float* __restrict__ neg,
                               float* __restrict__ acc) {
    const long long total = POS_LEN + NEG_TOTAL;
    const float wpos = 1.0f / (float)POS_LEN;
    const float wneg = 1.0f / (float)NEG_TOTAL;
    float v = 0.0f;
    long long stride = (long long)gridDim.x * blockDim.x;
    for (long long i = (long long)blockIdx.x * blockDim.x + threadIdx.x;
         i < total; i += stride) {
        if (i < POS_LEN) v += wpos * softplusf(-pos[i]);
        else             v += wneg * softplusf(neg[i - POS_LEN]);
    }
    v = block_reduce_sum(v);
    if (threadIdx.x == 0) atomicAdd(&acc[0], v);
}

// ---------------------------------------------------------------------------
// Kernel 2: regularization sum of squares (float4 streaming)
// ---------------------------------------------------------------------------
__global__ void reg_kernel(const float* __restrict__ ent,
                           const float* __restrict__ rel,
                           float* __restrict__ acc) {
    float v = 0.0f;
    long long stride = (long long)gridDim.x * blockDim.x;
    long long tid = (long long)blockIdx.x * blockDim.x + threadIdx.x;
    const float4* e4 = (const float4*)ent;
    for (long long i = tid; i < ENT_FLOATS / 4; i += stride) {
        float4 x = e4[i];
        v += x.x * x.x + x.y * x.y + x.z * x.z + x.w * x.w;
    }
    const float4* r4 = (const float4*)rel;
    for (long long i = tid; i < REL_FLOATS / 4; i += stride) {
        float4 x = r4[i];
        v += x.x * x.x + x.y * x.y + x.z * x.z + x.w * x.w;
    }
    v = block_reduce_sum(v);
    if (threadIdx.x == 0) atomicAdd(&acc[2], v);
}

// ---------------------------------------------------------------------------
// WMMA helpers. One wave32 computes 16 triple-product scores on the diagonal
// of a 16x16 f32 accumulator, chaining V_WMMA_F32_16X16X4_F32 over K=256.
//   A fragment (16x4 f32): lane L holds row j=L&15, cols k0+2*(L>=16)+{0,1}
//   B fragment (4x16 f32): lane L holds col n=L&15, rows k0+2*(L>=16)+{0,1}
// ---------------------------------------------------------------------------
__device__ __forceinline__ v8f wmma_step(const float* __restrict__ hp,
                                         const float* __restrict__ rp,
                                         const float* __restrict__ tp,
                                         int k, v8f c) {
    v2f a = (*(const v2f*)(hp + k)) * (*(const v2f*)(rp + k));
    v2f b = *(const v2f*)(tp + k);
    return __builtin_amdgcn_wmma_f32_16x16x4_f32(
        /*neg_a=*/false, a, /*neg_b=*/false, b,
        /*c_mod=*/(short)0, c, /*reuse_a=*/false, /*reuse_b=*/false);
}

// Two independent triple-product batches in one fused loop: the alternating
// c1/c2 WMMAs are independent, so the matrix pipe never stalls on the
// accumulator RAW chain, and 6 load streams stay in flight.
__device__ __forceinline__ void wave_score16_dual(
    const float* __restrict__ h1, const float* __restrict__ r1,
    const float* __restrict__ t1,
    const float* __restrict__ h2, const float* __restrict__ r2,
    const float* __restrict__ t2,
    int koff, v8f& c1o, v8f& c2o) {
    v8f c1 = {0.f, 0.f, 0.f, 0.f, 0.f, 0.f, 0.f, 0.f};
    v8f c2 = {0.f, 0.f, 0.f, 0.f, 0.f, 0.f, 0.f, 0.f};
#pragma unroll 2
    for (int k0 = 0; k0 < EMB_DIM; k0 += 4) {
        int k = k0 + koff;
        c1 = wmma_step(h1, r1, t1, k, c1);
        c2 = wmma_step(h2, r2, t2, k, c2);
    }
    c1o = c1;
    c2o = c2;
}

// Single score batch, 2-way ILP: even/odd K-chunks go to separate
// accumulators; diag(Ca + Cb) == wanted scores (same fragment layout).
__device__ __forceinline__ v8f wave_score16_single(
    const float* __restrict__ hp, const float* __restrict__ rp,
    const float* __restrict__ tp, int koff) {
    v8f ca = {0.f, 0.f, 0.f, 0.f, 0.f, 0.f, 0.f, 0.f};
    v8f cb = {0.f, 0.f, 0.f, 0.f, 0.f, 0.f, 0.f, 0.f};
#pragma unroll 2
    for (int k0 = 0; k0 < EMB_DIM; k0 += 8) {
        int k = k0 + koff;
        ca = wmma_step(hp, rp, tp, k, ca);
        cb = wmma_step(hp, rp, tp, k + 4, cb);
    }
    return ca + cb;
}

// Diagonal element ownership for the 16x16 f32 C layout:
//   i in [0,8):  lane i,     vgpr i
//   i in [8,16): lane i+16,  vgpr i-8
__device__ __forceinline__ bool diag_owner(int lane, int& idx, int& vreg) {
    if (lane < 8)        { idx = lane;      vreg = lane;      return true; }
    else if (lane >= 24) { idx = lane - 16; vreg = lane - 24; return true; }
    return false;
}

// ---------------------------------------------------------------------------
// Kernel 3: rule contributions -> atomic scatter into pi_grad
// ---------------------------------------------------------------------------
__global__ void rule_kernel(const float* __restrict__ ent,
                            const float* __restrict__ rel,
                            const int* __restrict__ p1,
                            const int* __restrict__ p2,
                            const float* __restrict__ conf,
                            const int* __restrict__ tnum,
                            const int* __restrict__ uid,
                            float* __restrict__ pi_grad) {
    int wave = (blockIdx.x * blockDim.x + threadIdx.x) >> 5;
    int lane = threadIdx.x & 31;
    int base = wave * 16;
    if (base >= N_RULES) return;              // wave-uniform: EXEC stays all-1s

    int j = lane & 15;
    long long t = base + j;                   // lanes L and L+16 share triple j
    const float* h1 = ent + (long long)p1[3 * t + 0] * EMB_DIM;
    const float* r1 = rel + (long long)p1[3 * t + 1] * EMB_DIM;
    const float* t1 = ent + (long long)p1[3 * t + 2] * EMB_DIM;
    const float* h2 = ent + (long long)p2[3 * t + 0] * EMB_DIM;
    const float* r2 = rel + (long long)p2[3 * t + 1] * EMB_DIM;
    const float* t2 = ent + (long long)p2[3 * t + 2] * EMB_DIM;

    // warm the gathered rows (global_prefetch_b8; L2-resident thereafter)
    __builtin_prefetch(h1, 0, 3);
    __builtin_prefetch(t1, 0, 3);
    __builtin_prefetch(h2, 0, 3);
    __builtin_prefetch(t2, 0, 3);

    int koff = (lane >> 4) * 2;
    v8f c1, c2;
    wave_score16_dual(h1, r1, t1, h2, r2, t2, koff, c1, c2);

    int idx, vreg;
    if (diag_owner(lane, idx, vreg)) {
        int ri = base + idx;
        float s1 = sigmoidf(c1[vreg]);
        float s2 = sigmoidf(c2[vreg]);
        float contrib = conf[ri] * s1 * (tnum[ri] == 3 ? s2 : 1.0f);
        atomicAdd(&pi_grad[uid[ri]], contrib);   // SLACKNESS_PENALTY == 1.0
    }
}

// ---------------------------------------------------------------------------
// Kernel 4: unlabel loss (WMMA score + clip target + softplus loss)
// ---------------------------------------------------------------------------
__global__ void unlabel_kernel(const float* __restrict__ ent,
                               const float* __restrict__ rel,
                               const int* __restrict__ utrip,
                               const float* __restrict__ pi_grad,
                               float* __restrict__ acc) {
    int wave = (blockIdx.x * blockDim.x + threadIdx.x) >> 5;
    int lane = threadIdx.x & 31;
    int base = wave * 16;
    if (base >= N_UNLABEL) return;

    int j = lane & 15;
    long long t = base + j;
    const float* hp = ent + (long long)utrip[3 * t + 0] * EMB_DIM;
    const float* rp = rel + (long long)utrip[3 * t + 1] * EMB_DIM;
    const float* tp = ent + (long long)utrip[3 * t + 2] * EMB_DIM;
    __builtin_prefetch(hp, 0, 3);
    __builtin_prefetch(tp, 0, 3);

    int koff = (lane >> 4) * 2;
    v8f c = wave_score16_single(hp, rp, tp, koff);

    int idx, vreg;
    if (diag_owner(lane, idx, vreg)) {
        int ui = base + idx;
        float su  = c[vreg];
        float p   = sigmoidf(su);
        float tgt = fminf(fmaxf(p + pi_grad[ui], 0.0f), 1.0f);
        float term = tgt * softplusf(-su) + (1.0f - tgt) * softplusf(su);
        atomicAdd(&acc[1], term * (1.0f / (float)N_UNLABEL));
    }
}

// ---------------------------------------------------------------------------
// Kernel 5: combine partials into the scalar output
// ---------------------------------------------------------------------------
__global__ void finalize_kernel(const float* __restrict__ acc,
                                float* __restrict__ out) {
    if (threadIdx.x == 0 && blockIdx.x == 0)
        out[0] = acc[0] + acc[1] + WDECAY * acc[2];
}

// ---------------------------------------------------------------------------
extern "C" void kernel_launch(void* const* d_in, const int* in_sizes, int n_in,
                              void* d_out, int out_size, void* d_ws, size_t ws_size,
                              hipStream_t stream) {
    const float* pos   = (const float*)d_in[0];   // (32768, 1)
    const float* neg   = (const float*)d_in[1];   // (32768, 64)
    const float* conf  = (const float*)d_in[2];   // (262144,)
    const float* ent   = (const float*)d_in[3];   // (100000, 256)
    const float* rel   = (const float*)d_in[4];   // (500, 256)
    const int*   p1    = (const int*)d_in[5];     // (262144, 3)
    const int*   p2    = (const int*)d_in[6];     // (262144, 3)
    const int*   tnum  = (const int*)d_in[7];     // (262144,)
    const int*   uid   = (const int*)d_in[8];     // (262144,)
    const int*   utrip = (const int*)d_in[9];     // (131072, 3)
    float* out = (float*)d_out;

    // workspace: acc[0..7] then pi_grad[131072]
    float* acc     = (float*)d_ws;
    float* pi_grad = acc + 8;

    zero_ws_kernel<<<512, 256, 0, stream>>>(acc, pi_grad, N_UNLABEL);
    pos_neg_kernel<<<2048, 256, 0, stream>>>(pos, neg, acc);
    reg_kernel<<<2048, 256, 0, stream>>>(ent, rel, acc);
    // 262144 rules / 16 per wave = 16384 waves; 8 waves/block -> 2048 blocks
    rule_kernel<<<2048, 256, 0, stream>>>(ent, rel, p1, p2, conf, tnum, uid, pi_grad);
    // 131072 / 16 = 8192 waves -> 1024 blocks
    unlabel_kernel<<<1024, 256, 0, stream>>>(ent, rel, utrip, pi_grad, acc);
    finalize_kernel<<<1, 32, 0, stream>>>(acc, out);
}